// GCN_12300786336289
// MI455X (gfx1250) — compile-verified
//
#include <hip/hip_runtime.h>
#include <hip/hip_bf16.h>

typedef __attribute__((ext_vector_type(2))) float v2f;
typedef __attribute__((ext_vector_type(8))) float v8f;

#define D_FEAT 64

// ---------------------------------------------------------------------------
// Kernel 1: zero the output accumulator and the degree-count scratch.
// ---------------------------------------------------------------------------
__global__ void gcn_init(float* __restrict__ out, unsigned* __restrict__ degc,
                         int n_out, int n_deg) {
    int stride = gridDim.x * blockDim.x;
    for (int i = blockIdx.x * blockDim.x + threadIdx.x; i < n_out; i += stride)
        out[i] = 0.0f;
    for (int i = blockIdx.x * blockDim.x + threadIdx.x; i < n_deg; i += stride)
        degc[i] = 0u;
}

// ---------------------------------------------------------------------------
// Kernel 2: degree histograms (u32 atomics, native).
// degc[0..N)  = out-degree counts, degc[N..2N) = in-degree counts.
// ---------------------------------------------------------------------------
__global__ void gcn_degrees(unsigned* __restrict__ degc,
                            const int* __restrict__ src,
                            const int* __restrict__ dst,
                            int n_edges, int n_nodes) {
    int gid = blockIdx.x * blockDim.x + threadIdx.x;
    if (gid < n_edges) {
        atomicAdd(&degc[src[gid]], 1u);
        atomicAdd(&degc[n_nodes + dst[gid]], 1u);
    }
}

// ---------------------------------------------------------------------------
// Kernel 3: in-place convert counts -> rsqrt(max(count,1)) as float bits.
// ---------------------------------------------------------------------------
__global__ void gcn_rsqrt(unsigned* __restrict__ deg, int n) {
    int gid = blockIdx.x * blockDim.x + threadIdx.x;
    if (gid < n) {
        unsigned c = deg[gid];
        float x = (c == 0u) ? 1.0f : (float)c;
        deg[gid] = __float_as_uint(rsqrtf(x));
    }
}

// ---------------------------------------------------------------------------
// Kernel 4: edge aggregation.  One wave32 per edge, 2 columns per lane.
// out[dst] += feat[src] * rdeg_out[src]   (fp32 fire-and-forget atomics; the
// 25.6MB node matrix is L2-resident on MI455X's 192MB L2).
// ---------------------------------------------------------------------------
__global__ void gcn_aggregate(float* __restrict__ out,
                              const float* __restrict__ feat,
                              const int* __restrict__ src,
                              const int* __restrict__ dst,
                              const float* __restrict__ rdeg_out,
                              int n_edges) {
    long gid = (long)blockIdx.x * blockDim.x + threadIdx.x;
    int e = (int)(gid >> 5);
    if (e >= n_edges) return;
    int lane = threadIdx.x & 31;

    int s = src[e];
    int d = dst[e];
    float sc = rdeg_out[s];

    const float2 f = *(const float2*)(feat + (size_t)s * D_FEAT + lane * 2);
    float* o = out + (size_t)d * D_FEAT + lane * 2;
    __hip_atomic_fetch_add(o + 0, f.x * sc, __ATOMIC_RELAXED, __HIP_MEMORY_SCOPE_AGENT);
    __hip_atomic_fetch_add(o + 1, f.y * sc, __ATOMIC_RELAXED, __HIP_MEMORY_SCOPE_AGENT);
}

// ---------------------------------------------------------------------------
// Kernel 5: fused scale+GEMM+bias with fp32 WMMA (V_WMMA_F32_16X16X4_F32).
//   out[r,:] = (rdeg_in[r] * agg[r,:]) @ W + bias       (agg aliases out)
// One wave per 16-row stripe; computes all 4 column tiles so every read of
// its rows precedes every write -> safe in-place.
// A layout (16x4 f32): lane l holds M=l&15, K=2*(l>>4)+{0,1}.
// B layout (4x16 f32): lane l holds N=l&15, K=2*(l>>4)+{0,1}.
// C/D: vgpr v holds M=v+8*(l>>4), N=l&15.
// ---------------------------------------------------------------------------
__global__ void gcn_gemm_wmma(float* out,
                              const float* __restrict__ W,
                              const float* __restrict__ bias,
                              const float* __restrict__ rdeg_in,
                              int n_tiles, int n_rows) {
    int wave = (blockIdx.x * blockDim.x + threadIdx.x) >> 5;
    if (wave >= n_tiles) return;                 // wave-uniform: EXEC stays all-1s
    int lane = threadIdx.x & 31;
    int half = lane >> 4;                        // 0 or 1
    int idx  = lane & 15;

    int row0 = wave * 16;
    int arow = row0 + idx;
    int arow_c = (arow < n_rows) ? arow : (n_rows - 1);
    float s = rdeg_in[arow_c];
    const float* Arow = out + (size_t)arow_c * D_FEAT;

    v8f c0 = {}, c1 = {}, c2 = {}, c3 = {};

#pragma unroll
    for (int k0 = 0; k0 < D_FEAT; k0 += 4) {
        int ka = k0 + 2 * half;
        v2f a;
        a.x = Arow[ka + 0] * s;
        a.y = Arow[ka + 1] * s;

        const float* B0 = W + (size_t)(ka + 0) * D_FEAT;
        const float* B1 = W + (size_t)(ka + 1) * D_FEAT;
        v2f b0, b1, b2, b3;
        b0.x = B0[idx +  0];  b0.y = B1[idx +  0];
        b1.x = B0[idx + 16];  b1.y = B1[idx + 16];
        b2.x = B0[idx + 32];  b2.y = B1[idx + 32];
        b3.x = B0[idx + 48];  b3.y = B1[idx + 48];

        c0 = __builtin_amdgcn_wmma_f32_16x16x4_f32(false, a, false, b0, (short)0, c0, false, false);
        c1 = __builtin_amdgcn_wmma_f32_16x16x4_f32(false, a, false, b1, (short)0, c1, false, false);
        c2 = __builtin_amdgcn_wmma_f32_16x16x4_f32(false, a, false, b2, (short)0, c2, false, false);
        c3 = __builtin_amdgcn_wmma_f32_16x16x4_f32(false, a, false, b3, (short)0, c3, false, false);
    }

    float bb0 = bias[idx +  0];
    float bb1 = bias[idx + 16];
    float bb2 = bias[idx + 32];
    float bb3 = bias[idx + 48];

#pragma unroll
    for (int v = 0; v < 8; ++v) {
        int m = v + 8 * half;
        int r = row0 + m;
        if (r < n_rows) {
            float* orow = out + (size_t)r * D_FEAT;
            orow[idx +  0] = c0[v] + bb0;
            orow[idx + 16] = c1[v] + bb1;
            orow[idx + 32] = c2[v] + bb2;
            orow[idx + 48] = c3[v] + bb3;
        }
    }
}

// ---------------------------------------------------------------------------
extern "C" void kernel_launch(void* const* d_in, const int* in_sizes, int n_in,
                              void* d_out, int out_size, void* d_ws, size_t ws_size,
                              hipStream_t stream) {
    const float* feat = (const float*)d_in[0];   // [N, 64]
    const float* W    = (const float*)d_in[1];   // [64, 64]
    const float* bias = (const float*)d_in[2];   // [64]
    const int*   src  = (const int*)d_in[3];     // [E]
    const int*   dst  = (const int*)d_in[4];     // [E]

    const int n_nodes = in_sizes[0] / D_FEAT;
    const int n_edges = in_sizes[3];

    float*    out  = (float*)d_out;              // doubles as aggregation buffer
    unsigned* degc = (unsigned*)d_ws;            // [2*N] u32 counts -> f32 rsqrt

    // 1) zero accumulator + degree counters
    gcn_init<<<2048, 256, 0, stream>>>(out, degc, n_nodes * D_FEAT, 2 * n_nodes);

    // 2) degree histograms
    gcn_degrees<<<(n_edges + 255) / 256, 256, 0, stream>>>(degc, src, dst,
                                                           n_edges, n_nodes);

    // 3) counts -> deg^{-1/2}
    gcn_rsqrt<<<(2 * n_nodes + 255) / 256, 256, 0, stream>>>(degc, 2 * n_nodes);

    // 4) edge scatter-add of pre-normalized features (one wave per edge)
    long agg_threads = (long)n_edges * 32;
    int  agg_blocks  = (int)((agg_threads + 255) / 256);
    gcn_aggregate<<<agg_blocks, 256, 0, stream>>>(out, feat, src, dst,
                                                  (const float*)degc, n_edges);

    // 5) fused post-scale + GEMM + bias via fp32 WMMA
    int n_tiles = (n_nodes + 15) / 16;
    gcn_gemm_wmma<<<(n_tiles + 7) / 8, 256, 0, stream>>>(
        out, W, bias, (const float*)degc + n_nodes, n_tiles, n_nodes);
}